// ExactScoreNetwork_74646531604676
// MI455X (gfx1250) — compile-verified
//
#include <hip/hip_runtime.h>
#include <cstdint>

// ---------------------------------------------------------------------------
// ExactScoreNetwork score kernel for MI455X (gfx1250).
//   out[b,d] = sqrt(1-e_b) * sum_k softmax_k(lp[b,k]) * diff/v
//   lp[b,k]  = -0.5*sum_d diff^2/v - 128*ln2*sum_d log2(v) + ln w_k  (+const dropped)
//   v = 1 + e_b*(s^2-1),  diff = x - a_b*c,  a=exp(-Beta/2), e=a^2
// Design: centers/stds TDM-DMA'd once per block into padded LDS (row stride
// 268 floats -> conflict-free), WMMA f32 16x16x4 performs both the d-reduction
// (ones-matrix trick) and the k-reduction (coeff-vector times diff/v matrix).
// ---------------------------------------------------------------------------

typedef float  v2f  __attribute__((ext_vector_type(2)));
typedef float  v8f  __attribute__((ext_vector_type(8)));
typedef unsigned int u32x4 __attribute__((ext_vector_type(4)));
typedef int    i32x4 __attribute__((ext_vector_type(4)));
typedef int    i32x8 __attribute__((ext_vector_type(8)));

#define K_N   128
#define D_N   256
#define ROWF  268                    // padded LDS row stride in floats (256+12)
#define C_BYTES (K_N * ROWF * 4)     // 137216 bytes per matrix
#define WAVE_SCRATCH 1536            // 256 x-floats + 128 coeff floats per wave
#define SMEM_BYTES (2 * C_BYTES + 8 * WAVE_SCRATCH)   // 286720 <= 320KB

__device__ __forceinline__ v8f wmma4(v2f a, v2f b, v8f c) {
  return __builtin_amdgcn_wmma_f32_16x16x4_f32(false, a, false, b, (short)0, c,
                                               false, false);
}

// TDM: DMA a row-major [128 x 256] f32 tensor from global into LDS, inserting
// 12 DWORDs of padding after every 256 DWORDs (one row) -> LDS row stride 268.
__device__ __forceinline__ void tdm_load_padded(const float* gsrc, uint32_t lds_off) {
  uint64_t ga = (uint64_t)(uintptr_t)gsrc;
  u32x4 g0;
  g0[0] = 1u;                                     // count=1, user descriptor
  g0[1] = lds_off;                                // LDS byte address
  g0[2] = (uint32_t)ga;                           // global addr [31:0]
  g0[3] = (uint32_t)((ga >> 32) & 0x01FFFFFFu)    // global addr [56:32]
          | (2u << 30);                           // type = 2 ("image")
  i32x8 g1;
  g1[0] = (int)((2u << 16)     // data_size = 4 bytes
              | (1u << 20)     // pad_enable
              | (7u << 22)     // pad_interval: 256 DWORDs (one 1KB row)
              | (11u << 25));  // pad_amount: 12 DWORDs
  g1[1] = (int)(256u << 16);                      // tensor_dim0 = 256 (lo16)
  g1[2] = (int)(128u << 16);                      // dim0 hi=0 | tensor_dim1=128 (lo16)
  g1[3] = (int)(256u << 16);                      // dim1 hi=0 | tile_dim0 = 256
  g1[4] = (int)(128u);                            // tile_dim1 = 128, tile_dim2 = 0
  g1[5] = 256;                                    // tensor_dim0_stride lo32 = 256
  g1[6] = 0;                                      // dim0_stride hi | dim1_stride lo
  g1[7] = 0;
  i32x4 g2 = {0, 0, 0, 0};
  i32x4 g3 = {0, 0, 0, 0};
#if defined(__clang_major__) && (__clang_major__ >= 23)
  i32x8 g4 = {0, 0, 0, 0, 0, 0, 0, 0};
  __builtin_amdgcn_tensor_load_to_lds(g0, g1, g2, g3, g4, 0);
#else
  __builtin_amdgcn_tensor_load_to_lds(g0, g1, g2, g3, 0);
#endif
}

__global__ __launch_bounds__(256, 1)
void exact_score_kernel(const float* __restrict__ x,
                        const float* __restrict__ t,
                        const float* __restrict__ centers,
                        const float* __restrict__ stds,
                        const float* __restrict__ weights,
                        float* __restrict__ out, int nB) {
  extern __shared__ char smem[];
  float* cL = (float*)smem;               // centers, [K][ROWF]
  float* sL = (float*)(smem + C_BYTES);   // stds -> transformed to s^2-1

  const int tid  = threadIdx.x;
  const int wave = tid >> 5;
  const int lane = tid & 31;
  const int n    = lane & 15;             // WMMA column / row-in-lane index
  const int h    = lane >> 4;             // half-wave select

  float* xw = (float*)(smem + 2 * C_BYTES + wave * WAVE_SCRATCH);  // 256 x
  float* cw = xw + D_N;                                            // 128 coeff

  // --- one-time per-block: TDM both matrices into padded LDS ---------------
  if (tid == 0) {
    tdm_load_padded(centers, (uint32_t)(uintptr_t)cL);
    tdm_load_padded(stds,    (uint32_t)(uintptr_t)sL);
    __builtin_amdgcn_s_wait_tensorcnt(0);
  }
  __syncthreads();
  // in-place transform: s -> s^2 - 1
  for (int i = tid; i < K_N * D_N; i += 256) {
    int r = i >> 8, col = i & 255;
    float s = sL[r * ROWF + col];
    sL[r * ROWF + col] = __builtin_fmaf(s, s, -1.0f);
  }
  __syncthreads();

  const float CL = 88.72283911167300f;    // 128 * ln2

  for (int b = blockIdx.x * 8 + wave; b < nB; b += (int)gridDim.x * 8) {
    const float* xrow = x + (size_t)b * D_N;
    __builtin_prefetch(xrow, 0, 3);
    // stage this sample's x row in per-wave LDS (read 2-wide in pass 1, scalar in pass 2)
    for (int j = lane; j < D_N; j += 32) xw[j] = xrow[j];

    const float tb   = t[b];
    const float Bt   = __builtin_fmaf(9.95f * tb, tb, 0.1f * tb);
    const float a    = __expf(-0.5f * Bt);
    const float e    = a * a;
    const float negA = -a;

    // ---- pass 1: lp[k] via WMMA ones-reduction over d ---------------------
    // B element (row kk=2h+v, col n) = contribution(k = k0+n, d = d0+2h+v)
    float lpArr[8];
    for (int kb = 0; kb < 8; ++kb) {
      const int k = kb * 16 + n;
      const float* crow = cL + k * ROWF;
      const float* srow = sL + k * ROWF;
      v8f acc = {0.f, 0.f, 0.f, 0.f, 0.f, 0.f, 0.f, 0.f};
      const v2f ones = {1.0f, 1.0f};
      for (int d0 = 0; d0 < D_N; d0 += 4) {
        const int d = d0 + 2 * h;
        v2f c2 = *(const v2f*)(crow + d);      // ds_load_b64, conflict-free
        v2f s2 = *(const v2f*)(srow + d);
        v2f x2 = *(const v2f*)(xw + d);
        v2f bm;
#pragma unroll
        for (int v = 0; v < 2; ++v) {
          float vv = __builtin_fmaf(e, s2[v], 1.0f);   // variance
          float rc = __builtin_amdgcn_rcpf(vv);
          float l2 = __log2f(vv);
          float df = __builtin_fmaf(negA, c2[v], x2[v]);
          float q  = df * df * rc;
          bm[v] = __builtin_fmaf(-CL, l2, -0.5f * q);
        }
        acc = wmma4(ones, bm, acc);            // column sums accumulate lp
      }
      lpArr[kb] = acc[0] + __logf(weights[k]); // lane n holds lp for k=kb*16+n
    }

    // ---- softmax over 128 logits (distributed over 16 lanes, halves dup) --
    float mx = lpArr[0];
#pragma unroll
    for (int kb = 1; kb < 8; ++kb) mx = fmaxf(mx, lpArr[kb]);
    for (int off = 1; off <= 8; off <<= 1) mx = fmaxf(mx, __shfl_xor(mx, off, 32));
    float p[8], zl = 0.f;
#pragma unroll
    for (int kb = 0; kb < 8; ++kb) { p[kb] = __expf(lpArr[kb] - mx); zl += p[kb]; }
    float Z = zl;
    for (int off = 1; off <= 8; off <<= 1) Z += __shfl_xor(Z, off, 32);
    const float invZ = __builtin_amdgcn_rcpf(Z);
    if (lane < 16) {
#pragma unroll
      for (int kb = 0; kb < 8; ++kb) cw[kb * 16 + n] = p[kb] * invZ;
    }

    // ---- pass 2: grad[d] = sum_k coeff_k * diff/v via WMMA ----------------
    // A element (m, kk=2h+v) = coeff[k0+kk] (replicated over m)
    // B element (row kk, col n) = diff/v at (k = k0+kk, d = d0+n)
    const float scale = __builtin_sqrtf(1.0f - e);
    float* orow = out + (size_t)b * D_N;
    for (int d0 = 0; d0 < D_N; d0 += 16) {
      const int d = d0 + n;
      const float xd = xw[d];
      v8f acc = {0.f, 0.f, 0.f, 0.f, 0.f, 0.f, 0.f, 0.f};
      for (int k0 = 0; k0 < K_N; k0 += 4) {
        const int kk = k0 + 2 * h;
        v2f a2 = *(const v2f*)(cw + kk);       // coeff pair for this lane
        v2f bm;
#pragma unroll
        for (int v = 0; v < 2; ++v) {
          const int k = kk + v;
          float cc = cL[k * ROWF + d];
          float ss = sL[k * ROWF + d];
          float vv = __builtin_fmaf(e, ss, 1.0f);
          float rc = __builtin_amdgcn_rcpf(vv);
          float df = __builtin_fmaf(negA, cc, xd);
          bm[v] = df * rc;
        }
        acc = wmma4(a2, bm, acc);
      }
      if (lane < 16) orow[d] = scale * acc[0]; // grad replicated over rows
    }
  }
}

extern "C" void kernel_launch(void* const* d_in, const int* in_sizes, int n_in,
                              void* d_out, int out_size, void* d_ws, size_t ws_size,
                              hipStream_t stream) {
  const float* x       = (const float*)d_in[0];
  const float* t       = (const float*)d_in[1];
  const float* centers = (const float*)d_in[2];
  const float* stds    = (const float*)d_in[3];
  const float* weights = (const float*)d_in[4];
  float* out = (float*)d_out;

  const int nB = in_sizes[0] / D_N;                 // 4096
  const int blocks = (nB + 7) / 8;                  // one b per wave, 8 waves/block

  hipFuncSetAttribute((const void*)exact_score_kernel,
                      hipFuncAttributeMaxDynamicSharedMemorySize, SMEM_BYTES);
  exact_score_kernel<<<dim3(blocks), dim3(256), SMEM_BYTES, stream>>>(
      x, t, centers, stds, weights, out, nB);
}